// Standard_76811195121796
// MI455X (gfx1250) — compile-verified
//
#include <hip/hip_runtime.h>

// Lorenz Lyapunov-exponent batch integrator for MI455X (gfx1250, wave32).
// One lane = one trajectory. All state (x, Q, lya) lives in VGPRs for the
// whole 100-step loop; only 12 B read + 24 B written per lane. Compute-bound
// on f32 VALU; uses v_log_f32 / v_rcp_f32 / v_rsq_f32 hardware transcendentals
// instead of IEEE div/sqrt expansions. WMMA intentionally unused: the only
// matmul is a batched 3x3 with per-lane operands, which cannot be expressed
// as a shared-K 16x16 tile without >20x efficiency loss.

#define SIGMA_C 10.0f
#define RHO_C   28.0f
#define BETA_C  2.6666666666666665f   // 8/3
#define DT_C    0.01f
#define T_STEPS 100
#define HALF_LN2 0.34657359027997264f // 0.5 * ln(2)

__device__ __forceinline__ void lorenz_rhs(float X, float Y, float Z,
                                           float& dX, float& dY, float& dZ) {
    dX = SIGMA_C * (Y - X);
    dY = X * (RHO_C - Z) - Y;
    dZ = X * Y - BETA_C * Z;
}

__global__ __launch_bounds__(256) void lorenz_lyap_kernel(
    const float* __restrict__ x0,   // (3, B) row-major
    const float* __restrict__ ts,   // (T)
    float* __restrict__ out,        // lya (3,B) then x (3,B)
    int B)
{
    const int b = blockIdx.x * blockDim.x + threadIdx.x;
    if (b >= B) return;

    // gfx1250 prefetch path (global_prefetch_b8); overlaps with setup, free.
    __builtin_prefetch(x0 + b, 0, 0);
    __builtin_prefetch(x0 + 2 * B + b, 0, 0);

    float X = x0[b];
    float Y = x0[B + b];
    float Z = x0[2 * B + b];

    // Q = I (columns are the orthonormal frame), lya = 0
    float q00 = 1.f, q01 = 0.f, q02 = 0.f;
    float q10 = 0.f, q11 = 1.f, q12 = 0.f;
    float q20 = 0.f, q21 = 0.f, q22 = 1.f;
    float l0 = 0.f, l1 = 0.f, l2 = 0.f;

#pragma unroll 1
    for (int k = 0; k < T_STEPS; ++k) {
        const float t = ts[k];  // uniform -> scalar load

        // ---- RK4 (reference quirk: k4 evaluated at x + dt*k2) ----
        float k1x, k1y, k1z, k2x, k2y, k2z, k3x, k3y, k3z, k4x, k4y, k4z;
        lorenz_rhs(X, Y, Z, k1x, k1y, k1z);
        lorenz_rhs(X + 0.5f * DT_C * k1x, Y + 0.5f * DT_C * k1y,
                   Z + 0.5f * DT_C * k1z, k2x, k2y, k2z);
        lorenz_rhs(X + 0.5f * DT_C * k2x, Y + 0.5f * DT_C * k2y,
                   Z + 0.5f * DT_C * k2z, k3x, k3y, k3z);
        lorenz_rhs(X + DT_C * k2x, Y + DT_C * k2y,
                   Z + DT_C * k2z, k4x, k4y, k4z);
        const float w = DT_C * (1.0f / 6.0f);
        X += w * (k1x + 2.f * k2x + 2.f * k3x + k4x);
        Y += w * (k1y + 2.f * k2y + 2.f * k3y + k4y);
        Z += w * (k1z + 2.f * k2z + 2.f * k3z + k4z);

        // ---- J = I + dt * Df(x_new) ----
        const float j00 = 1.f - DT_C * SIGMA_C;
        const float j01 = DT_C * SIGMA_C;
        const float j10 = DT_C * (RHO_C - Z);
        const float j11 = 1.f - DT_C;
        const float j12 = -DT_C * X;
        const float j20 = DT_C * Y;
        const float j21 = DT_C * X;
        const float j22 = 1.f - DT_C * BETA_C;

        // ---- N = J * Q (j02 == 0 folded out) ----
        const float n00 = j00 * q00 + j01 * q10;
        const float n01 = j00 * q01 + j01 * q11;
        const float n02 = j00 * q02 + j01 * q12;
        const float n10 = j10 * q00 + j11 * q10 + j12 * q20;
        const float n11 = j10 * q01 + j11 * q11 + j12 * q21;
        const float n12 = j10 * q02 + j11 * q12 + j12 * q22;
        const float n20 = j20 * q00 + j21 * q10 + j22 * q20;
        const float n21 = j20 * q01 + j21 * q11 + j22 * q21;
        const float n22 = j20 * q02 + j21 * q12 + j22 * q22;

        // ---- Classical Gram-Schmidt on columns (matches reference) ----
        // b0 = col0
        const float b00 = n00, b10 = n10, b20 = n20;
        const float beta0 = b00 * b00 + b10 * b10 + b20 * b20;
        const float r0 = __builtin_amdgcn_rcpf(beta0);   // v_rcp_f32
        // b1 = col1 - (b0 . col1)/beta0 * b0
        const float d01 = b00 * n01 + b10 * n11 + b20 * n21;
        const float c1 = d01 * r0;
        const float b01 = n01 - c1 * b00;
        const float b11 = n11 - c1 * b10;
        const float b21 = n21 - c1 * b20;
        const float beta1 = b01 * b01 + b11 * b11 + b21 * b21;
        const float r1 = __builtin_amdgcn_rcpf(beta1);
        // b2 = col2 - (b0 . col2)/beta0 * b0 - (b1 . col2)/beta1 * b1
        const float d02 = b00 * n02 + b10 * n12 + b20 * n22;
        const float d12 = b01 * n02 + b11 * n12 + b21 * n22;
        const float c20 = d02 * r0;
        const float c21 = d12 * r1;
        const float b02 = n02 - c20 * b00 - c21 * b01;
        const float b12 = n12 - c20 * b10 - c21 * b11;
        const float b22 = n22 - c20 * b20 - c21 * b21;
        const float beta2 = b02 * b02 + b12 * b12 + b22 * b22;

        // ---- Normalize columns: scale by rsqrt(||b||^2); no sqrt needed ----
        const float s0 = __builtin_amdgcn_rsqf(beta0);   // v_rsq_f32
        const float s1 = __builtin_amdgcn_rsqf(beta1);
        const float s2 = __builtin_amdgcn_rsqf(beta2);
        q00 = b00 * s0; q10 = b10 * s0; q20 = b20 * s0;
        q01 = b01 * s1; q11 = b11 * s1; q21 = b21 * s1;
        q02 = b02 * s2; q12 = b12 * s2; q22 = b22 * s2;

        // ---- lya = (lya*t + log(norm)) / (t+dt);  log(norm)=0.5*ln(beta) ----
        const float inv = __builtin_amdgcn_rcpf(t + DT_C);
        l0 = (l0 * t + HALF_LN2 * __builtin_amdgcn_logf(beta0)) * inv; // v_log_f32
        l1 = (l1 * t + HALF_LN2 * __builtin_amdgcn_logf(beta1)) * inv;
        l2 = (l2 * t + HALF_LN2 * __builtin_amdgcn_logf(beta2)) * inv;
    }

    // Output: lya (3,B) flat, then x (3,B) flat — coalesced stores.
    out[b]         = l0;
    out[B + b]     = l1;
    out[2 * B + b] = l2;
    out[3 * B + b] = X;
    out[4 * B + b] = Y;
    out[5 * B + b] = Z;
}

extern "C" void kernel_launch(void* const* d_in, const int* in_sizes, int n_in,
                              void* d_out, int out_size, void* d_ws, size_t ws_size,
                              hipStream_t stream) {
    const float* x0 = (const float*)d_in[0];   // (3, B) float32
    const float* ts = (const float*)d_in[1];   // (T)    float32
    float* out = (float*)d_out;                // 6*B float32
    const int B = in_sizes[0] / 3;

    const int block = 256;                     // 8 wave32 waves / block
    const int grid = (B + block - 1) / block;  // 1024 blocks at B=262144
    lorenz_lyap_kernel<<<grid, block, 0, stream>>>(x0, ts, out, B);
}